// LatentDynamicsExtrapolator_14070312862145
// MI455X (gfx1250) — compile-verified
//
#include <hip/hip_runtime.h>
#include <hip/hip_bf16.h>
#include <math.h>

typedef __attribute__((ext_vector_type(2))) float v2f;
typedef __attribute__((ext_vector_type(8))) float v8f;

__device__ __forceinline__ float gelu_f(float x) {
    return 0.5f * x * (1.0f + erff(x * 0.70710678118654752440f));
}

// ---------------------------------------------------------------------------
// Generic f32 WMMA GEMM: C[M,N] = A[M,K] * B[K,N]  (+ epilogue)
// BTRANS=0: B[k][n] = Bp[k*ldb + n]   (row-major B)
// BTRANS=1: B[k][n] = Bp[n*ldb + k]   (B = W^T for torch-style [out,in] weights)
// EPI=0: C[m*ldc+n] = acc
// EPI=1: C[m*ldc+n] = gelu(acc + bias[n])
// EPI=2: decoder remap: m = n_node*16 + b -> C[b*524288 + n_node*512 + n] = acc+bias[n]
// Block: 128 threads (4 waves). Block tile 64x64, K-step 32, per-wave 16x64.
// A tile LDS: [64][36] (padded stride -> conflict-free, float4-aligned stores).
// B tile LDS: K-pair interleaved [16][64][2] -> B fragment is one ds_load_b64.
// Register double-buffering: next K-tile's global loads are issued right after
// the barrier so they overlap the current tile's WMMA work (no serialized
// load->wait->ds_store chains).
// ---------------------------------------------------------------------------
#define GT_M 64
#define GT_N 64
#define GT_K 32
#define A_STRIDE 36

template <int BTRANS, int EPI>
__global__ __launch_bounds__(128) void gemm_wmma(
    const float* __restrict__ A, int lda,
    const float* __restrict__ Bp, int ldb,
    float* __restrict__ C, int ldc,
    int M, int N, int K,
    const float* __restrict__ bias)
{
    __shared__ float As[GT_M * A_STRIDE];   // [64][36]
    __shared__ float Bs[GT_K * GT_N];       // [16][64][2] pair-interleaved

    const int tid  = threadIdx.x;
    const int wave = tid >> 5;
    const int lane = tid & 31;
    const int l16  = lane & 15;
    const int half = lane >> 4;
    const int mt   = blockIdx.y * GT_M;
    const int nt   = blockIdx.x * GT_N;

    v8f acc[4];
#pragma unroll
    for (int t = 0; t < 4; t++)
#pragma unroll
        for (int v = 0; v < 8; v++) acc[t][v] = 0.0f;

    const int mrow = wave * 16 + l16;  // this lane's A row within block tile

    float4 ra[4], rb[4];

    auto load_tile = [&](int kt) {
#pragma unroll
        for (int i = 0; i < 4; i++) {
            int idx = i * 128 + tid;        // float4 index 0..511
            int row = idx >> 3;             // 0..63
            int c4  = idx & 7;              // 0..7
            ra[i] = *(const float4*)(A + (size_t)(mt + row) * lda + kt + c4 * 4);
        }
        if (BTRANS) {
#pragma unroll
            for (int i = 0; i < 4; i++) {
                int idx = i * 128 + tid;    // float4 index along K
                int n   = idx >> 3;
                int c4  = idx & 7;
                rb[i] = *(const float4*)(Bp + (size_t)(nt + n) * ldb + kt + c4 * 4);
            }
        } else {
#pragma unroll
            for (int i = 0; i < 4; i++) {
                int idx = i * 128 + tid;
                int row = idx >> 4;         // k row 0..31
                int c4  = idx & 15;         // n quarter 0..15
                rb[i] = *(const float4*)(Bp + (size_t)(kt + row) * ldb + nt + c4 * 4);
            }
        }
    };

    auto store_tile = [&]() {
#pragma unroll
        for (int i = 0; i < 4; i++) {
            int idx = i * 128 + tid;
            int row = idx >> 3;
            int c4  = idx & 7;
            float* dst = &As[row * A_STRIDE + c4 * 4];
            dst[0] = ra[i].x; dst[1] = ra[i].y; dst[2] = ra[i].z; dst[3] = ra[i].w;
        }
        // Bs element (k,n) lives at (k>>1)*128 + n*2 + (k&1)
        if (BTRANS) {
#pragma unroll
            for (int i = 0; i < 4; i++) {
                int idx = i * 128 + tid;
                int n   = idx >> 3;
                int c4  = idx & 7;
                int k0  = c4 * 4;           // even
                v2f lo; lo.x = rb[i].x; lo.y = rb[i].y;
                v2f hi; hi.x = rb[i].z; hi.y = rb[i].w;
                *(v2f*)&Bs[(k0 >> 1) * 128 + n * 2]       = lo;  // (k0, k0+1)
                *(v2f*)&Bs[((k0 >> 1) + 1) * 128 + n * 2] = hi;  // (k0+2, k0+3)
            }
        } else {
#pragma unroll
            for (int i = 0; i < 4; i++) {
                int idx = i * 128 + tid;
                int row = idx >> 4;
                int c4  = idx & 15;
                int kb  = (row >> 1) * 128 + (row & 1);
                int n0  = c4 * 4;
                Bs[kb + (n0 + 0) * 2] = rb[i].x;
                Bs[kb + (n0 + 1) * 2] = rb[i].y;
                Bs[kb + (n0 + 2) * 2] = rb[i].z;
                Bs[kb + (n0 + 3) * 2] = rb[i].w;
            }
        }
    };

    load_tile(0);

    for (int kt = 0; kt < K; kt += GT_K) {
        store_tile();
        __syncthreads();
        if (kt + GT_K < K) load_tile(kt + GT_K);  // overlaps WMMA below

        // ---- WMMA inner loop: 8 k-chunks x 4 N-tiles (A frag reused 4x) ----
#pragma unroll
        for (int k0 = 0; k0 < GT_K; k0 += 4) {
            int kk = k0 + half * 2;         // ISA frag layout: half-wave K offset
            v2f a = *(const v2f*)&As[mrow * A_STRIDE + kk];
#pragma unroll
            for (int t = 0; t < 4; t++) {
                int n = t * 16 + l16;
                v2f b = *(const v2f*)&Bs[(kk >> 1) * 128 + n * 2];
                acc[t] = __builtin_amdgcn_wmma_f32_16x16x4_f32(
                    false, a, false, b, (short)0, acc[t], false, false);
            }
        }
        __syncthreads();
    }

    // ---- epilogue (C frag: VGPR v -> row v + 8*half) ----
#pragma unroll
    for (int t = 0; t < 4; t++) {
        int n = nt + t * 16 + l16;
#pragma unroll
        for (int v = 0; v < 8; v++) {
            int m = mt + wave * 16 + v + 8 * half;
            float val = acc[t][v];
            if (EPI == 1) {
                C[(size_t)m * ldc + n] = gelu_f(val + bias[n]);
            } else if (EPI == 2) {
                val += bias[n];
                int bb = m & 15;
                int nn = m >> 4;
                C[((size_t)bb << 19) + (size_t)nn * 512 + n] = val; // [16][1024][512]
            } else {
                C[(size_t)m * ldc + n] = val;
            }
        }
    }
}

// ---------------------------------------------------------------------------
// Encoder: z0 = Lin2(gelu(Lin1(x[:, -1]))) -> packed Zw[r = n*16+b][32]
// ---------------------------------------------------------------------------
__global__ __launch_bounds__(256) void encoder_kernel(
    const float* __restrict__ x, const float* __restrict__ w1,
    const float* __restrict__ b1, const float* __restrict__ w2,
    const float* __restrict__ b2, float* __restrict__ Zw)
{
    __shared__ float sw1[64], sb1[64], sw2[32 * 64], sb2[32];
    int tid = threadIdx.x;
    if (tid < 64) { sw1[tid] = w1[tid]; sb1[tid] = b1[tid]; }
    for (int i = tid; i < 2048; i += 256) sw2[i] = w2[i];
    if (tid < 32) sb2[tid] = b2[tid];
    __syncthreads();

    int r = blockIdx.x * 256 + tid;       // r = n*16 + b
    int b = r & 15, n = r >> 4;
    float xv = x[(size_t)b * 294912 + 287 * 1024 + n];  // x[b, T-1, n]
    float h[64];
#pragma unroll
    for (int j = 0; j < 64; j++) h[j] = gelu_f(sw1[j] * xv + sb1[j]);
#pragma unroll
    for (int d = 0; d < 32; d++) {
        float acc = sb2[d];
#pragma unroll
        for (int j = 0; j < 64; j++) acc += sw2[d * 64 + j] * h[j];
        Zw[(size_t)r * 32 + d] = acc;
    }
}

// ---------------------------------------------------------------------------
// Per-step time vectors: t_feat[32], h[32]=gelu(tm_w1@t_feat+b), tbias[64]
// (tbias folds the t_feat half of tmlp_w1 plus tmlp_b1)
// ---------------------------------------------------------------------------
__global__ __launch_bounds__(64) void tvec_kernel(
    int step, const int* __restrict__ gap,
    const float* __restrict__ te_w, const float* __restrict__ te_b,
    const float* __restrict__ tm_w1, const float* __restrict__ tm_b1,
    const float* __restrict__ tmlp_w1, const float* __restrict__ tmlp_b1,
    float* __restrict__ tf_out, float* __restrict__ h_out, float* __restrict__ tb_out)
{
    __shared__ float tf[32];
    int tid = threadIdx.x;
    float dt = (float)gap[0] * 8760.0f;   // (gap*365*24*12)/12
    float t  = dt * (float)step;
    const float per[4] = {1.0f, 288.0f, 2016.0f, 105120.0f};
    float feats[8];
#pragma unroll
    for (int p = 0; p < 4; p++) {
        float ang = 6.28318530717958647692f * t / per[p];
        feats[2 * p]     = cosf(ang);
        feats[2 * p + 1] = sinf(ang);
    }
    if (tid < 32) {
        float acc = te_b[tid];
#pragma unroll
        for (int i = 0; i < 8; i++) acc += te_w[tid * 8 + i] * feats[i];
        tf[tid] = acc;
        tf_out[tid] = acc;
    }
    __syncthreads();
    if (tid < 32) {
        float acc = tm_b1[tid];
#pragma unroll
        for (int i = 0; i < 32; i++) acc += tm_w1[tid * 32 + i] * tf[i];
        h_out[tid] = gelu_f(acc);
    }
    {
        float acc = tmlp_b1[tid];
#pragma unroll
        for (int i = 0; i < 32; i++) acc += tmlp_w1[tid * 64 + 32 + i] * tf[i];
        tb_out[tid] = acc;
    }
}

// ---------------------------------------------------------------------------
// adj kernel: one block per node-row n. mod is batch-invariant (t scalar/step):
// mod[n,m] = tm_w2[n*1024+m] . h + tm_b2[n*1024+m];  adj = softmax(base*mod)
// ---------------------------------------------------------------------------
__global__ __launch_bounds__(256) void adj_kernel(
    const float* __restrict__ base, const float* __restrict__ tm_w2,
    const float* __restrict__ tm_b2, const float* __restrict__ hvec,
    float* __restrict__ adjw)
{
    __shared__ float red[256];
    __shared__ float h[32];
    int n = blockIdx.x;
    int tid = threadIdx.x;
    if (tid < 32) h[tid] = hvec[tid];
    __syncthreads();

    float s[4];
    float lmax = -INFINITY;
#pragma unroll
    for (int j = 0; j < 4; j++) {
        int m = tid + j * 256;
        const float* wrow = tm_w2 + ((size_t)n * 1024 + m) * 32;
        float dot = 0.0f;
#pragma unroll
        for (int k = 0; k < 32; k += 4) {
            float4 w = *(const float4*)(wrow + k);
            dot += w.x * h[k] + w.y * h[k + 1] + w.z * h[k + 2] + w.w * h[k + 3];
        }
        float sv = base[(size_t)n * 1024 + m] * (dot + tm_b2[(size_t)n * 1024 + m]);
        s[j] = sv;
        lmax = fmaxf(lmax, sv);
    }
    red[tid] = lmax; __syncthreads();
    for (int off = 128; off > 0; off >>= 1) {
        if (tid < off) red[tid] = fmaxf(red[tid], red[tid + off]);
        __syncthreads();
    }
    float mx = red[0]; __syncthreads();
    float lsum = 0.0f;
#pragma unroll
    for (int j = 0; j < 4; j++) { s[j] = expf(s[j] - mx); lsum += s[j]; }
    red[tid] = lsum; __syncthreads();
    for (int off = 128; off > 0; off >>= 1) {
        if (tid < off) red[tid] += red[tid + off];
        __syncthreads();
    }
    float inv = 1.0f / red[0];
#pragma unroll
    for (int j = 0; j < 4; j++) adjw[(size_t)n * 1024 + tid + j * 256] = s[j] * inv;
}

// ---------------------------------------------------------------------------
// Fused ODE-step epilogue: graph-proj GELU + time MLP + LayerNorm + Euler.
// One thread per packed row r=(n*16+b).
// ---------------------------------------------------------------------------
__global__ __launch_bounds__(128) void update_kernel(
    const float* __restrict__ C1, float* __restrict__ Zw,
    const float* __restrict__ gp_w, const float* __restrict__ gp_b,
    const float* __restrict__ tmlp_w1, const float* __restrict__ tbias,
    const float* __restrict__ tmlp_w2, const float* __restrict__ tmlp_b2,
    const float* __restrict__ ln_w, const float* __restrict__ ln_b,
    const int* __restrict__ gap)
{
    __shared__ float sgpw[32 * 32], sw1[64 * 32], sw2[32 * 64];
    __shared__ float sgpb[32], stb[64], sb2[32], slw[32], slb[32];
    int tid = threadIdx.x;
    for (int i = tid; i < 1024; i += 128) sgpw[i] = gp_w[i];
    for (int i = tid; i < 2048; i += 128) { int j = i >> 5, k = i & 31; sw1[i] = tmlp_w1[j * 64 + k]; }
    for (int i = tid; i < 2048; i += 128) sw2[i] = tmlp_w2[i];
    if (tid < 32) { sgpb[tid] = gp_b[tid]; sb2[tid] = tmlp_b2[tid]; slw[tid] = ln_w[tid]; slb[tid] = ln_b[tid]; }
    if (tid < 64) stb[tid] = tbias[tid];
    __syncthreads();

    float dt = (float)gap[0] * 8760.0f;
    int r = blockIdx.x * 128 + tid;
    const float* c1r = C1 + (size_t)r * 32;
    float* zp = Zw + (size_t)r * 32;
    float a1[32], zr[32];
#pragma unroll
    for (int d = 0; d < 32; d++) { a1[d] = c1r[d]; zr[d] = zp[d]; }

    float u[64];
#pragma unroll
    for (int j = 0; j < 64; j++) {
        float acc = stb[j];
#pragma unroll
        for (int i = 0; i < 32; i++) acc += sw1[j * 32 + i] * zr[i];
        u[j] = gelu_f(acc);
    }
    float h2[32]; float mu = 0.0f;
#pragma unroll
    for (int o = 0; o < 32; o++) {
        float sp = sgpb[o];
#pragma unroll
        for (int i = 0; i < 32; i++) sp += sgpw[o * 32 + i] * a1[i];
        sp = gelu_f(sp);
        float tm = sb2[o];
#pragma unroll
        for (int j = 0; j < 64; j++) tm += sw2[o * 64 + j] * u[j];
        h2[o] = sp + tm;
        mu += h2[o];
    }
    mu *= (1.0f / 32.0f);
    float var = 0.0f;
#pragma unroll
    for (int o = 0; o < 32; o++) { float dd = h2[o] - mu; var += dd * dd; }
    var *= (1.0f / 32.0f);
    float rinv = rsqrtf(var + 1e-5f);
#pragma unroll
    for (int o = 0; o < 32; o++) {
        float zn = (h2[o] - mu) * rinv * slw[o] + slb[o];
        zp[o] = zr[o] + dt * zn;
    }
}

// ---------------------------------------------------------------------------
extern "C" void kernel_launch(void* const* d_in, const int* in_sizes, int n_in,
                              void* d_out, int out_size, void* d_ws, size_t ws_size,
                              hipStream_t stream)
{
    (void)in_sizes; (void)n_in; (void)out_size; (void)ws_size;
    const float* x       = (const float*)d_in[0];
    const float* enc_w1  = (const float*)d_in[1];
    const float* enc_b1  = (const float*)d_in[2];
    const float* enc_w2  = (const float*)d_in[3];
    const float* enc_b2  = (const float*)d_in[4];
    const float* te_w    = (const float*)d_in[5];
    const float* te_b    = (const float*)d_in[6];
    const float* emb1    = (const float*)d_in[7];
    const float* emb2    = (const float*)d_in[8];
    const float* tm_w1   = (const float*)d_in[9];
    const float* tm_b1   = (const float*)d_in[10];
    const float* tm_w2   = (const float*)d_in[11];
    const float* tm_b2   = (const float*)d_in[12];
    const float* gp_w    = (const float*)d_in[13];
    const float* gp_b    = (const float*)d_in[14];
    const float* tmlp_w1 = (const float*)d_in[15];
    const float* tmlp_b1 = (const float*)d_in[16];
    const float* tmlp_w2 = (const float*)d_in[17];
    const float* tmlp_b2 = (const float*)d_in[18];
    const float* ln_w    = (const float*)d_in[19];
    const float* ln_b    = (const float*)d_in[20];
    const float* dec_w1  = (const float*)d_in[21];
    const float* dec_b1  = (const float*)d_in[22];
    const float* dec_w2  = (const float*)d_in[23];
    const float* dec_b2  = (const float*)d_in[24];
    const int*   gap     = (const int*)d_in[25];

    float* ws   = (float*)d_ws;
    float* Zw   = ws;                       // [16384][32] == [1024][512] view (2 MB)
    float* base = Zw + 524288;              // [1024][1024] (4 MB)
    float* adjw = base + 1048576;           // [1024][1024] (4 MB)
    float* C1   = adjw + 1048576;           // [16384][32] (2 MB)
    float* tf   = C1 + 524288;              // [32]
    float* hh   = tf + 32;                  // [32]
    float* tb   = hh + 32;                  // [64]
    float* H    = tb + 64;                  // [16384][1024] (67 MB)

    encoder_kernel<<<64, 256, 0, stream>>>(x, enc_w1, enc_b1, enc_w2, enc_b2, Zw);

    // base = emb1 @ emb2^T   (M=1024, N=1024, K=32)
    gemm_wmma<1, 0><<<dim3(16, 16), 128, 0, stream>>>(
        emb1, 32, emb2, 32, base, 1024, 1024, 1024, 32, nullptr);

    for (int s = 0; s < 12; s++) {
        tvec_kernel<<<1, 64, 0, stream>>>(s, gap, te_w, te_b, tm_w1, tm_b1,
                                          tmlp_w1, tmlp_b1, tf, hh, tb);
        adj_kernel<<<1024, 256, 0, stream>>>(base, tm_w2, tm_b2, hh, adjw);
        // C1 = adj @ Zw   (M=1024, N=512, K=1024)
        gemm_wmma<0, 0><<<dim3(8, 16), 128, 0, stream>>>(
            adjw, 1024, Zw, 512, C1, 512, 1024, 512, 1024, nullptr);
        update_kernel<<<128, 128, 0, stream>>>(C1, Zw, gp_w, gp_b, tmlp_w1, tb,
                                               tmlp_w2, tmlp_b2, ln_w, ln_b, gap);
    }

    // H = gelu(Zw_flat @ dec_w1^T + b1)   (M=16384, N=1024, K=32)
    gemm_wmma<1, 1><<<dim3(16, 256), 128, 0, stream>>>(
        Zw, 32, dec_w1, 32, H, 1024, 16384, 1024, 32, dec_b1);
    // out = H @ dec_w2^T + b2, remapped to [16][1024][512]
    gemm_wmma<1, 2><<<dim3(8, 256), 128, 0, stream>>>(
        H, 1024, dec_w2, 1024, (float*)d_out, 512, 16384, 512, 1024, dec_b2);
}